// Reservoir_34488587387687
// MI455X (gfx1250) — compile-verified
//
#include <hip/hip_runtime.h>

typedef __attribute__((ext_vector_type(2))) float v2f;
typedef __attribute__((ext_vector_type(8))) float v8f;

#define DIM_RES 8192
#define DIM_IN  64
#define DIM_OUT 32
#define LEAK    0.01f   // DT / TAU = 0.1 / 10.0

// ---------------------------------------------------------------------------
// Phase 1: state = W@r + W_in@u + W_fb@y_prev ; r_new = r + LEAK*(tanh(state)-r)
//
// Each block: 16 rows of W. Each of the 8 waves: a 1024-wide K slice, consumed
// as 16x4 f32 WMMA tiles. A-operand (W tile) layout per ISA 7.12.2:
//   lane L (0..15):  row = L,     VGPR0 = W[row,k0+0], VGPR1 = W[row,k0+1]
//   lane L (16..31): row = L-16,  VGPR0 = W[row,k0+2], VGPR1 = W[row,k0+3]
// -> one global b64 load per lane streams the 16x4 tile exactly once.
// B is r broadcast across all 16 columns: B[k,j] = r[k] for all j, so
// D[i,j] = sum_k W[i,k] r[k] lands (replicated over j) in the D registers.
// 4 independent accumulators break the WMMA D->C dependency chain.
// ---------------------------------------------------------------------------
__global__ __launch_bounds__(256) void reservoir_update_kernel(
    const float* __restrict__ W,
    const float* __restrict__ Win,
    const float* __restrict__ Wfb,
    const float* __restrict__ r,
    const float* __restrict__ u,
    const float* __restrict__ yprev,
    float* __restrict__ rnew)
{
    __shared__ float red[8][16];

    const int tid  = threadIdx.x;
    const int wave = tid >> 5;       // 0..7
    const int lane = tid & 31;
    const int row  = blockIdx.x * 16 + (lane & 15);
    const int koff = (lane >> 4) << 1;            // 0 for lanes 0-15, 2 for 16-31
    const float* __restrict__ Wp = W + (size_t)row * DIM_RES + koff;
    const float* __restrict__ rp = r + koff;

    v8f c0 = {}, c1 = {}, c2 = {}, c3 = {};

    const int kbase = wave * (DIM_RES / 8);       // 1024-wide slice per wave
    #pragma unroll 2
    for (int k = kbase; k < kbase + DIM_RES / 8; k += 16) {
        v2f a0 = *(const v2f*)(Wp + k);
        v2f a1 = *(const v2f*)(Wp + k + 4);
        v2f a2 = *(const v2f*)(Wp + k + 8);
        v2f a3 = *(const v2f*)(Wp + k + 12);
        v2f b0 = *(const v2f*)(rp + k);
        v2f b1 = *(const v2f*)(rp + k + 4);
        v2f b2 = *(const v2f*)(rp + k + 8);
        v2f b3 = *(const v2f*)(rp + k + 12);
        c0 = __builtin_amdgcn_wmma_f32_16x16x4_f32(false, a0, false, b0, (short)0, c0, false, false);
        c1 = __builtin_amdgcn_wmma_f32_16x16x4_f32(false, a1, false, b1, (short)0, c1, false, false);
        c2 = __builtin_amdgcn_wmma_f32_16x16x4_f32(false, a2, false, b2, (short)0, c2, false, false);
        c3 = __builtin_amdgcn_wmma_f32_16x16x4_f32(false, a3, false, b3, (short)0, c3, false, false);
    }
    v8f c = (c0 + c1) + (c2 + c3);

    // D layout: VGPR v holds (row = v + 8*(lane/16), col = lane%16); all cols
    // equal (broadcast B). Lane 0 carries rows 0-7, lane 16 carries rows 8-15.
    if (lane == 0) {
        #pragma unroll
        for (int i = 0; i < 8; ++i) red[wave][i] = c[i];
    } else if (lane == 16) {
        #pragma unroll
        for (int i = 0; i < 8; ++i) red[wave][8 + i] = c[i];
    }
    __syncthreads();

    if (tid < 16) {
        float s = 0.0f;
        #pragma unroll
        for (int w2 = 0; w2 < 8; ++w2) s += red[w2][tid];

        const int grow = blockIdx.x * 16 + tid;
        const float* __restrict__ wi = Win + (size_t)grow * DIM_IN;
        #pragma unroll 8
        for (int j = 0; j < DIM_IN; ++j) s += wi[j] * u[j];
        const float* __restrict__ wf = Wfb + (size_t)grow * DIM_OUT;
        #pragma unroll 8
        for (int j = 0; j < DIM_OUT; ++j) s += wf[j] * yprev[j];

        const float rv = r[grow];
        rnew[grow] = rv + LEAK * (tanhf(s) - rv);
    }
}

// ---------------------------------------------------------------------------
// Phase 2: y = W_out @ r_new  (32 x 8192). One block per output row.
// ---------------------------------------------------------------------------
__global__ __launch_bounds__(256) void readout_kernel(
    const float* __restrict__ Wout,
    const float* __restrict__ rnew,
    float* __restrict__ out)
{
    __shared__ float red[256];
    const int b = blockIdx.x;
    const int t = threadIdx.x;

    const float* __restrict__ wr = Wout + (size_t)b * DIM_RES;
    float s = 0.0f;
    for (int k = t; k < DIM_RES; k += 256) s += wr[k] * rnew[k];
    red[t] = s;
    __syncthreads();
    #pragma unroll
    for (int st = 128; st > 0; st >>= 1) {
        if (t < st) red[t] += red[t + st];
        __syncthreads();
    }
    if (t == 0) out[b] = red[0];
}

extern "C" void kernel_launch(void* const* d_in, const int* in_sizes, int n_in,
                              void* d_out, int out_size, void* d_ws, size_t ws_size,
                              hipStream_t stream) {
    (void)in_sizes; (void)n_in; (void)out_size; (void)ws_size;
    // setup_inputs order: input_signal, W, W_in, W_fb, W_out, r, out_prev
    const float* u     = (const float*)d_in[0];
    const float* W     = (const float*)d_in[1];
    const float* Win   = (const float*)d_in[2];
    const float* Wfb   = (const float*)d_in[3];
    const float* Wout  = (const float*)d_in[4];
    const float* r     = (const float*)d_in[5];
    const float* yprev = (const float*)d_in[6];
    float* out  = (float*)d_out;
    float* rnew = (float*)d_ws;   // 8192 floats = 32 KB scratch

    reservoir_update_kernel<<<DIM_RES / 16, 256, 0, stream>>>(W, Win, Wfb, r, u, yprev, rnew);
    readout_kernel<<<DIM_OUT, 256, 0, stream>>>(Wout, rnew, out);
}